// SparseConv3DBlock_31473520345591
// MI455X (gfx1250) — compile-verified
//
#include <hip/hip_runtime.h>
#include <hip/hip_bf16.h>

#define N_PTS 200000
#define F_DIM 32
#define K_OFF 27
#define BN_EPS 1e-5f
#define TILES (N_PTS / 16)          // 12500 exactly
#define TILES_PER_WAVE 4
#define WT_STRIDE 34                // padded transposed-W row stride (dwords)

typedef __attribute__((ext_vector_type(2))) float v2f;
typedef __attribute__((ext_vector_type(8))) float v8f;
typedef __attribute__((ext_vector_type(4))) int   v4i;

// ---------------------------------------------------------------------------
// Kernel 0: zero fp64 stats, zero dummy h_pad row, out = bias broadcast
// ---------------------------------------------------------------------------
__global__ void spconv_init_kernel(float* __restrict__ out,
                                   const float* __restrict__ bias,
                                   float* __restrict__ hpad_dummy,
                                   double* __restrict__ stats) {
    size_t idx = (size_t)blockIdx.x * blockDim.x + threadIdx.x;
    if (idx < 64) stats[idx] = 0.0;
    if (idx < F_DIM) hpad_dummy[idx] = 0.0f;
    size_t stride = (size_t)gridDim.x * blockDim.x;
    for (size_t i = idx; i < (size_t)N_PTS * F_DIM; i += stride)
        out[i] = bias[i & (F_DIM - 1)];
}

// ---------------------------------------------------------------------------
// Kernel 1: per-channel sum / sumsq (fp64 accumulation, f64 atomics to ws)
// ---------------------------------------------------------------------------
#define STAT_ROWS 2048
__global__ __launch_bounds__(256) void spconv_stats_kernel(
        const float* __restrict__ feats, double* __restrict__ stats) {
    __shared__ double ssum[256];
    __shared__ double ssq[256];
    const int c = threadIdx.x & (F_DIM - 1);
    const int g = threadIdx.x >> 5;           // 0..7 row groups
    size_t start = (size_t)blockIdx.x * STAT_ROWS;
    size_t end   = start + STAT_ROWS;
    if (end > N_PTS) end = N_PTS;
    double s = 0.0, q = 0.0;
    for (size_t r = start + g; r < end; r += 8) {
        float x = feats[r * F_DIM + c];
        s += (double)x;
        q += (double)x * (double)x;
    }
    ssum[threadIdx.x] = s;
    ssq[threadIdx.x]  = q;
    __syncthreads();
    if (g == 0) {
        for (int j = 1; j < 8; ++j) {
            s += ssum[c + 32 * j];
            q += ssq[c + 32 * j];
        }
        __hip_atomic_fetch_add(&stats[c], s,
                               __ATOMIC_RELAXED, __HIP_MEMORY_SCOPE_AGENT);
        __hip_atomic_fetch_add(&stats[F_DIM + c], q,
                               __ATOMIC_RELAXED, __HIP_MEMORY_SCOPE_AGENT);
    }
}

// ---------------------------------------------------------------------------
// Kernel 2: BN (training-mode batch stats) + SiLU -> h_pad rows [0, N)
// ---------------------------------------------------------------------------
__global__ void spconv_bn_silu_kernel(const float* __restrict__ feats,
                                      const float* __restrict__ gamma,
                                      const float* __restrict__ beta,
                                      const double* __restrict__ stats,
                                      float* __restrict__ h_pad) {
    size_t i = (size_t)blockIdx.x * blockDim.x + threadIdx.x;
    if (i >= (size_t)N_PTS * F_DIM) return;
    int c = (int)(i & (F_DIM - 1));
    double m = stats[c] / (double)N_PTS;
    double v = stats[F_DIM + c] / (double)N_PTS - m * m;
    float h = (feats[i] - (float)m) * rsqrtf((float)v + BN_EPS) * gamma[c] + beta[c];
    h_pad[i] = h / (1.0f + __expf(-h));     // SiLU
}

// ---------------------------------------------------------------------------
// Kernel 3: gather -> fp32 WMMA (16x16x4) -> atomic scatter-add
// block = 256 (8 waves); wave w handles TILES_PER_WAVE consecutive 16-row
// tiles for offset k = blockIdx.y. W[k] is staged in LDS *transposed +
// padded* (WT[n*34 + kr] = W[kr][n]) so every B fragment is one aligned
// conflict-free ds_load_b64 straight into an even VGPR pair — no re-pairing
// movs in front of the WMMAs.
// ---------------------------------------------------------------------------
__global__ __launch_bounds__(256) void SparseConv3DBlock_31473520345591_kernel(
        const float* __restrict__ h_pad,        // (N_PTS+1, 32), row N_PTS = zeros
        const float* __restrict__ W,            // (27, 32, 32)
        const int*   __restrict__ in_map,       // (27, N_PTS), pad = N_PTS
        const int*   __restrict__ out_map,      // (27, N_PTS), pad = N_PTS
        float*       __restrict__ out) {        // (N_PTS, 32), pre-seeded with bias
    __shared__ float WT[F_DIM * WT_STRIDE];     // 4.4 KB, transposed W[k]
    const int k = blockIdx.y;
    const float* Wk = W + (size_t)k * F_DIM * F_DIM;
    for (int idx = threadIdx.x; idx < F_DIM * F_DIM; idx += blockDim.x) {
        const int kin = idx >> 5;               // K row    (0..31)
        const int n   = idx & (F_DIM - 1);      // out col  (0..31)
        WT[n * WT_STRIDE + kin] = Wk[idx];      // coalesced read, 32-bank store
    }
    __syncthreads();

    const int wave = threadIdx.x >> 5;
    const int lane = threadIdx.x & 31;
    const int half = lane >> 4;                 // 0: K pair {0,1}; 1: {2,3}
    const int mlo  = lane & 15;

    const float* bt0 = &WT[mlo * WT_STRIDE + 2 * half];          // N = mlo
    const float* bt1 = &WT[(mlo + 16) * WT_STRIDE + 2 * half];   // N = mlo+16

    const int tile0 = (blockIdx.x * 8 + wave) * TILES_PER_WAVE;
    const size_t kmap = (size_t)k * N_PTS;

#pragma unroll
    for (int j = 0; j < TILES_PER_WAVE; ++j) {
        const int tile = tile0 + j;             // wave-uniform
        if (tile >= TILES) break;               // wave-uniform -> EXEC stays full
        const int row_base = tile * 16;
        const size_t map_base = kmap + row_base;

        // A-row gather pointer for this lane (dummy row N_PTS holds zeros)
        const int irow = in_map[map_base + mlo];
        const float* arow = h_pad + (size_t)irow * F_DIM + 2 * half;

        // scatter rows for this lane-half: two b128 broadcast loads
        const int* omp = out_map + map_base + 8 * half;
        const v4i o0 = *(const v4i*)omp;        // m = 8*half + 0..3
        const v4i o1 = *(const v4i*)(omp + 4);  // m = 8*half + 4..7

        v8f acc0 = {};                          // N = 0..15
        v8f acc1 = {};                          // N = 16..31
#pragma unroll
        for (int c = 0; c < 8; ++c) {           // K chunks of 4 over F_IN=32
            v2f a  = *(const v2f*)(arow + 4 * c);     // A[m][4c+2h .. +1]
            v2f b0 = *(const v2f*)(bt0 + 4 * c);      // single ds_load_b64
            v2f b1 = *(const v2f*)(bt1 + 4 * c);
            acc0 = __builtin_amdgcn_wmma_f32_16x16x4_f32(
                       false, a, false, b0, (short)0, acc0, false, false);
            acc1 = __builtin_amdgcn_wmma_f32_16x16x4_f32(
                       false, a, false, b1, (short)0, acc1, false, false);
        }

        // D layout: VGPR r -> M = r + 8*half, N = mlo (+16 for acc1)
#pragma unroll
        for (int r = 0; r < 8; ++r) {
            const int orow = (r < 4) ? o0[r] : o1[r - 4];
            if (orow < N_PTS) {
                float* o = out + (size_t)orow * F_DIM + mlo;
                __hip_atomic_fetch_add(o, acc0[r],
                                       __ATOMIC_RELAXED, __HIP_MEMORY_SCOPE_AGENT);
                __hip_atomic_fetch_add(o + 16, acc1[r],
                                       __ATOMIC_RELAXED, __HIP_MEMORY_SCOPE_AGENT);
            }
        }
    }
}

// ---------------------------------------------------------------------------
extern "C" void kernel_launch(void* const* d_in, const int* in_sizes, int n_in,
                              void* d_out, int out_size, void* d_ws, size_t ws_size,
                              hipStream_t stream) {
    const float* feats   = (const float*)d_in[0];   // (N, 32)
    const float* gamma   = (const float*)d_in[1];   // (32,)
    const float* beta    = (const float*)d_in[2];   // (32,)
    const float* W       = (const float*)d_in[3];   // (27, 32, 32)
    const float* bias    = (const float*)d_in[4];   // (32,)
    const int*   in_map  = (const int*)d_in[5];     // (27, N)
    const int*   out_map = (const int*)d_in[6];     // (27, N)
    float* out = (float*)d_out;                     // (N, 32)

    // workspace layout: [0,512) fp64 stats (sum[32], sumsq[32]); then h_pad
    double* stats = (double*)d_ws;
    float*  h_pad = (float*)((char*)d_ws + 512);    // (N+1, 32)

    // 0: init out (bias), stats, dummy row
    spconv_init_kernel<<<1024, 256, 0, stream>>>(
        out, bias, h_pad + (size_t)N_PTS * F_DIM, stats);

    // 1: batch statistics
    int stat_blocks = (N_PTS + STAT_ROWS - 1) / STAT_ROWS;
    spconv_stats_kernel<<<stat_blocks, 256, 0, stream>>>(feats, stats);

    // 2: BN + SiLU
    int bn_blocks = (int)(((size_t)N_PTS * F_DIM + 255) / 256);
    spconv_bn_silu_kernel<<<bn_blocks, 256, 0, stream>>>(
        feats, gamma, beta, stats, h_pad);

    // 3: WMMA gather-GEMM-scatter over 27 offsets
    const int waves_per_block = 8;
    const int tiles_per_block = waves_per_block * TILES_PER_WAVE;   // 32
    dim3 grid((TILES + tiles_per_block - 1) / tiles_per_block, K_OFF);
    SparseConv3DBlock_31473520345591_kernel<<<grid, 256, 0, stream>>>(
        h_pad, W, in_map, out_map, out);
}